// ViewMorphing_62594853372362
// MI455X (gfx1250) — compile-verified
//
#include <hip/hip_runtime.h>
#include <hip/hip_bf16.h>

#define IMG_D   512
#define HW      (IMG_D * IMG_D)      // 262144 = 2^18
#define NB      16
#define BLK     256

typedef __attribute__((ext_vector_type(2))) float v2f;
typedef __attribute__((ext_vector_type(8))) float v8f;

#if defined(__has_builtin)
#  if __has_builtin(__builtin_amdgcn_wmma_f32_16x16x4_f32)
#    define HAVE_WMMA_F32_16X16X4 1
#  endif
#endif

// ---- CDNA5 async global -> LDS copy (ASYNCcnt path, bypasses VGPRs) -------
__device__ __forceinline__ void async_g2lds_b32(const float* g, unsigned lds_byte_addr) {
    // GV mode: global_load_async_to_lds_b32 vdst(lds addr), vaddr(64b), off
    asm volatile("global_load_async_to_lds_b32 %0, %1, off"
                 :: "v"(lds_byte_addr), "v"(g) : "memory");
}
__device__ __forceinline__ void wait_asynccnt0() {
    asm volatile("s_wait_asynccnt 0" ::: "memory");
}
// low 32 bits of a generic shared pointer == workgroup-relative LDS byte addr
__device__ __forceinline__ unsigned lds_addr_of(const void* p) {
    return (unsigned)(size_t)p;
}

// ---- 4-corner bilinear gather over 3 channel planes -----------------------
__device__ __forceinline__ void sample3(const float* __restrict__ base, // n-th image, 3*HW plane
                                        float qx, float qy, float o[3]) {
    float xf = floorf(qx), yf = floorf(qy);
    float xc = ceilf(qx),  yc = ceilf(qy);
    float wxf = 1.0f - (qx - xf);
    float wxc = 1.0f - (xc - qx);
    float wyf = 1.0f - (qy - yf);
    float wyc = 1.0f - (yc - qy);
    int ixf = (int)xf, ixc = (int)xc, iyf = (int)yf, iyc = (int)yc;
    int iFF = iyf + (ixf << 9);
    int iCF = iyf + (ixc << 9);
    int iFC = iyc + (ixf << 9);
    int iCC = iyc + (ixc << 9);
    float wFF = wxf * wyf, wCF = wxc * wyf, wFC = wxf * wyc, wCC = wxc * wyc;
#pragma unroll
    for (int c = 0; c < 3; ++c) {
        const float* pl = base + (size_t)c * HW;
        o[c] = wFF * pl[iFF] + wCF * pl[iCF] + wFC * pl[iFC] + wCC * pl[iCC];
    }
}

__global__ void __launch_bounds__(BLK)
vm_zero_ws(float* ws) { ws[0] = 0.0f; }

__global__ void __launch_bounds__(BLK)
vm_main(const float* __restrict__ im1, const float* __restrict__ im2,
        const float* __restrict__ C,   const float* __restrict__ M1,
        const float* __restrict__ M2,  float* __restrict__ out,
        float* __restrict__ ws, int atomic_mode) {
    __shared__ float sC0[BLK], sC1[BLK], sM1[BLK], sM2[BLK];
    __shared__ float spart[BLK];

    const int tid = threadIdx.x;
    const long long t = (long long)blockIdx.x * BLK + tid;
    const int n = (int)(t >> 18);          // HW = 2^18, BLK divides HW -> n uniform per block
    const int p = (int)(t & (HW - 1));

    // --- stage streaming operands via async DMA to LDS ---------------------
    const float* pC0 = C  + ((size_t)n * 2) * HW + p;
    const float* pC1 = pC0 + HW;
    const float* pM1 = M1 + (size_t)n * HW + p;
    const float* pM2 = M2 + (size_t)n * HW + p;
    async_g2lds_b32(pC0, lds_addr_of(&sC0[tid]));
    async_g2lds_b32(pC1, lds_addr_of(&sC1[tid]));
    async_g2lds_b32(pM1, lds_addr_of(&sM1[tid]));
    async_g2lds_b32(pM2, lds_addr_of(&sM2[tid]));

    // overlap DMA with address math
    const float q0 = (float)(p >> 9);      // x = flat / D
    const float q1 = (float)(p & 511);     // y = flat % D
    const float* b1 = im1 + (size_t)n * 3 * HW;
    const float* b2 = im2 + (size_t)n * 3 * HW;

    wait_asynccnt0();
    __syncthreads();

    const float c0 = sC0[tid], c1 = sC1[tid];
    const float m1 = sM1[tid], m2 = sM2[tid];

    float oob = 0.0f;
    float oA[3], oB[3];

    {   // view A: q + C, sample im1
        float rx = (q0 + c0) * 512.0f;
        float ry = (q1 + c1) * 512.0f;
        float qx = fminf(fmaxf(rx, 0.001f), 510.999f);
        float qy = fminf(fmaxf(ry, 0.001f), 510.999f);
        float dx = rx - qx, dy = ry - qy;
        oob += dx * dx + dy * dy;
        sample3(b1, qx, qy, oA);
    }
    {   // view B: q - C, sample im2
        float rx = (q0 - c0) * 512.0f;
        float ry = (q1 - c1) * 512.0f;
        float qx = fminf(fmaxf(rx, 0.001f), 510.999f);
        float qy = fminf(fmaxf(ry, 0.001f), 510.999f);
        float dx = rx - qx, dy = ry - qy;
        oob += dx * dx + dy * dy;
        sample3(b2, qx, qy, oB);
    }

    const float tm  = m1 + m2;
    const float m1n = m1 / tm;
    const float m2n = m2 / tm;
    float* ob = out + (size_t)n * 3 * HW + p;
    ob[0]        = m1n * oA[0] + m2n * oB[0];
    ob[HW]       = m1n * oA[1] + m2n * oB[1];
    ob[2 * HW]   = m1n * oA[2] + m2n * oB[2];

    // --- per-block OOB reduction -------------------------------------------
    spart[tid] = oob;
    __syncthreads();
    if (tid < 32) {                        // exactly wave 0, EXEC all-ones (wave32)
        float v;
#if defined(HAVE_WMMA_F32_16X16X4_F32) || defined(HAVE_WMMA_F32_16X16X4)
        // ones-matmul reduction: D = 1s(16x4) x B(4x16) + C -> every row = colsums(B);
        // any bijective packing of 64 values into B's 2 VGPRs sums exactly.
        v2f a; a.x = 1.0f; a.y = 1.0f;
        v8f acc = {};
#pragma unroll
        for (int j = 0; j < 4; ++j) {
            v2f b;
            b.x = spart[j * 64 + tid];
            b.y = spart[j * 64 + 32 + tid];
            acc = __builtin_amdgcn_wmma_f32_16x16x4_f32(
                false, a, false, b, (short)0, acc, false, false);
        }
        v = acc[0];                        // row M=0 (lanes 0-15) and M=8 (lanes 16-31)
#pragma unroll
        for (int m = 16; m >= 1; m >>= 1) v += __shfl_xor(v, m, 32);
        v *= 0.5f;                         // rows 0 and 8 each carry the full colsum set
#else
        v = 0.0f;
#pragma unroll
        for (int k = 0; k < 8; ++k) v += spart[tid + 32 * k];
#pragma unroll
        for (int m = 16; m >= 1; m >>= 1) v += __shfl_xor(v, m, 32);
#endif
        if (tid == 0) {
            if (atomic_mode) atomicAdd(&ws[0], v);
            else             ws[blockIdx.x] = v;
        }
    }
}

__global__ void __launch_bounds__(BLK)
vm_finalize(const float* __restrict__ ws, int nblocks, int atomic_mode,
            float* __restrict__ out_scalar) {
    // total / (N*2*HW) / D^2 * 1e-4
    const float scale = (1.0f / 8388608.0f) * (1.0f / 262144.0f) * 1.0e-4f;
    if (atomic_mode) {
        if (threadIdx.x == 0) *out_scalar = ws[0] * scale;
        return;
    }
    __shared__ float s[BLK];
    float v = 0.0f;
    for (int i = threadIdx.x; i < nblocks; i += BLK) v += ws[i];   // fixed order
    s[threadIdx.x] = v;
    __syncthreads();
    for (int st = BLK / 2; st > 0; st >>= 1) {
        if (threadIdx.x < st) s[threadIdx.x] += s[threadIdx.x + st];
        __syncthreads();
    }
    if (threadIdx.x == 0) *out_scalar = s[0] * scale;
}

extern "C" void kernel_launch(void* const* d_in, const int* in_sizes, int n_in,
                              void* d_out, int out_size, void* d_ws, size_t ws_size,
                              hipStream_t stream) {
    const float* im1 = (const float*)d_in[0];
    const float* im2 = (const float*)d_in[1];
    const float* C   = (const float*)d_in[2];
    const float* M1  = (const float*)d_in[3];
    const float* M2  = (const float*)d_in[4];
    float* out = (float*)d_out;
    float* ws  = (float*)d_ws;

    const int total   = NB * HW;           // 4,194,304 threads
    const int nblocks = total / BLK;       // 16384
    const int atomic_mode = (ws_size < (size_t)nblocks * sizeof(float)) ? 1 : 0;

    if (atomic_mode)
        vm_zero_ws<<<1, BLK, 0, stream>>>(ws);

    vm_main<<<nblocks, BLK, 0, stream>>>(im1, im2, C, M1, M2, out, ws, atomic_mode);
    vm_finalize<<<1, BLK, 0, stream>>>(ws, nblocks, atomic_mode,
                                       out + (size_t)out_size - 1);
}